// PointNetPreprocessor_2963527435033
// MI455X (gfx1250) — compile-verified
//
#include <hip/hip_runtime.h>
#include <hip/hip_bf16.h>

#define FPS_NUM   512
#define GROUP_NUM 32
#define NPTS      32768
#define NBATCH    8
#define RADIUS2   0.04f
#define BIG       1e10f
#define CHUNK     256

typedef __attribute__((ext_vector_type(2))) float v2f;
typedef __attribute__((ext_vector_type(8))) float v8f;

// ---------------------------------------------------------------------------
// Kernel 1: Farthest Point Sampling.
// One block per batch, 1024 threads (32 wave32 waves). Running min-distance
// kept in registers (32 per thread). Per-iteration argmax: intra-wave
// shfl_xor reduce, then a single 32-entry LDS stage reduced by wave 0.
// Emits centroid xyz for step s *before* the distance update (matches scan).
// ---------------------------------------------------------------------------
__global__ __launch_bounds__(1024) void fps_kernel(
    const float* __restrict__ xyz,
    const int*   __restrict__ farthest_init,
    float*       __restrict__ out_centroids /* [B, FPS_NUM, 3] */)
{
    const int b    = blockIdx.x;
    const int tid  = threadIdx.x;
    const int lane = tid & 31;
    const int wid  = tid >> 5;                 // 0..31
    const float* p = xyz + (size_t)b * NPTS * 3;

    __shared__ float s_val[32];
    __shared__ int   s_idx[32];
    __shared__ float s_c[3];
    __shared__ int   s_far;

    float dist[NPTS / 1024];
    #pragma unroll
    for (int j = 0; j < NPTS / 1024; ++j) dist[j] = BIG;

    if (tid == 0) s_far = farthest_init[b];
    __syncthreads();

    for (int s = 0; s < FPS_NUM; ++s) {
        if (tid == 0) {
            const int f = s_far;
            const float cx = p[f * 3 + 0];
            const float cy = p[f * 3 + 1];
            const float cz = p[f * 3 + 2];
            s_c[0] = cx; s_c[1] = cy; s_c[2] = cz;
            float* oc = out_centroids + ((size_t)b * FPS_NUM + s) * 3;
            oc[0] = cx; oc[1] = cy; oc[2] = cz;
        }
        __syncthreads();
        const float cx = s_c[0], cy = s_c[1], cz = s_c[2];

        float best = -1.0f;
        int   bi   = tid;
        #pragma unroll
        for (int j = 0; j < NPTS / 1024; ++j) {
            const int   i  = tid + j * 1024;
            const float dx = p[i * 3 + 0] - cx;
            const float dy = p[i * 3 + 1] - cy;
            const float dz = p[i * 3 + 2] - cz;
            float d = dx * dx + dy * dy + dz * dz;
            d = fminf(dist[j], d);
            dist[j] = d;
            if (d > best) { best = d; bi = i; }   // strict > keeps lowest index
        }

        // wave32 argmax reduce (lower index wins ties, as jnp.argmax)
        #pragma unroll
        for (int off = 16; off > 0; off >>= 1) {
            const float ov = __shfl_xor(best, off, 32);
            const int   oi = __shfl_xor(bi,   off, 32);
            if (ov > best || (ov == best && oi < bi)) { best = ov; bi = oi; }
        }
        if (lane == 0) { s_val[wid] = best; s_idx[wid] = bi; }
        __syncthreads();
        if (wid == 0) {
            best = s_val[lane];
            bi   = s_idx[lane];
            #pragma unroll
            for (int off = 16; off > 0; off >>= 1) {
                const float ov = __shfl_xor(best, off, 32);
                const int   oi = __shfl_xor(bi,   off, 32);
                if (ov > best || (ov == best && oi < bi)) { best = ov; bi = oi; }
            }
            if (lane == 0) s_far = bi;
        }
        __syncthreads();
    }
}

// ---------------------------------------------------------------------------
// Kernel 2: Ball query + grouping, WMMA-accelerated.
// Block = 256 threads (8 waves) handles a tile of 16 centroids of one batch.
// Distances computed per 256-point chunk via V_WMMA_F32_16X16X4_F32:
//   dist = |c|^2 + |p|^2 - 2 * (c . p),  K = (x,y,z,0)
// Operands staged in 4-row LDS arrays (x,y,z,0) so lane-half operand loads
// are unconditional (no exec save/restore in the hot loop). Each wave
// produces two 16x16 distance tiles into LDS, then ballot-compaction
// selection for two centroids. Early-exits when all 16 centroids are full.
// ---------------------------------------------------------------------------
__global__ __launch_bounds__(256) void ballquery_group_kernel(
    const float* __restrict__ xyz,
    const float* __restrict__ centroids_xyz, /* [B, FPS_NUM, 3] */
    float*       __restrict__ out            /* [B, FPS_NUM, G+1, 3] */)
{
    const int tile = blockIdx.x;               // 0 .. FPS_NUM/16 - 1
    const int b    = blockIdx.y;
    const int tid  = threadIdx.x;
    const int lane = tid & 31;
    const int wid  = tid >> 5;                 // 0..7
    const float* p = xyz + (size_t)b * NPTS * 3;

    __shared__ float s_p[4][CHUNK];            // rows: x, y, z, 0 (K-pad)
    __shared__ float s_pn[CHUNK];
    __shared__ float s_c4[4][16];              // rows: x, y, z, 0 (K-pad)
    __shared__ float s_cn[16];
    __shared__ float s_dist[16][CHUNK];
    __shared__ int   s_grp[16][GROUP_NUM];
    __shared__ int   s_cnt[16];
    __shared__ int   s_done;

    if (tid < 16) {
        const float* c = centroids_xyz + ((size_t)b * FPS_NUM + tile * 16 + tid) * 3;
        const float x = c[0], y = c[1], z = c[2];
        s_c4[0][tid] = x; s_c4[1][tid] = y; s_c4[2][tid] = z; s_c4[3][tid] = 0.0f;
        s_cn[tid] = x * x + y * y + z * z;
        s_cnt[tid] = 0;
    }
    s_p[3][tid] = 0.0f;                        // zero K-pad row, never rewritten
    __syncthreads();

    // A-matrix (16x4, centroids) is invariant across chunks.
    // ISA layout: lanes 0-15 hold K=0,1 ; lanes 16-31 hold K=2,3.
    const int half = lane >> 4;
    const int mn   = lane & 15;
    v2f amat;
    amat.x = s_c4[2 * half + 0][mn];
    amat.y = s_c4[2 * half + 1][mn];

    for (int chunk = 0; chunk < NPTS / CHUNK; ++chunk) {
        const int base = chunk * CHUNK;

        // stage chunk of points + squared norms into LDS
        {
            const int   i = base + tid;
            const float x = p[i * 3 + 0];
            const float y = p[i * 3 + 1];
            const float z = p[i * 3 + 2];
            s_p[0][tid] = x; s_p[1][tid] = y; s_p[2][tid] = z;
            s_pn[tid] = x * x + y * y + z * z;
        }
        // prefetch next chunk into the cache hierarchy (global_prefetch_b8)
        if (chunk + 1 < NPTS / CHUNK) {
            __builtin_prefetch(p + (size_t)(base + CHUNK + tid) * 3, 0, 1);
        }
        __syncthreads();

        // WMMA phase: 16 n-tiles of 16 points; wave w does tiles w and w+8.
        #pragma unroll
        for (int t = 0; t < 2; ++t) {
            const int ntile = wid + t * 8;
            const int nloc  = ntile * 16 + mn;     // local point index
            v2f bmat;
            bmat.x = s_p[2 * half + 0][nloc];
            bmat.y = s_p[2 * half + 1][nloc];
            v8f acc = {};
            acc = __builtin_amdgcn_wmma_f32_16x16x4_f32(
                false, amat, false, bmat, (short)0, acc, false, false);
            // C layout: VGPR r, lane-half h -> row m = r + 8*h, col n = lane&15
            #pragma unroll
            for (int r = 0; r < 8; ++r) {
                const int m = r + 8 * half;
                s_dist[m][nloc] = s_cn[m] + s_pn[nloc] - 2.0f * acc[r];
            }
        }
        __syncthreads();

        // Selection: wave w handles centroids 2w and 2w+1 via ballot-compact.
        #pragma unroll
        for (int cc = 0; cc < 2; ++cc) {
            const int c = wid * 2 + cc;
            int cnt = s_cnt[c];                    // wave-uniform
            for (int g = 0; g < CHUNK / 32 && cnt < GROUP_NUM; ++g) {
                const int   il = g * 32 + lane;
                const float d  = s_dist[c][il];
                const bool  ok = (d <= RADIUS2);
                const unsigned int mask = (unsigned int)__ballot(ok);
                if (ok) {
                    const int pos = cnt + __popc(mask & ((1u << lane) - 1u));
                    if (pos < GROUP_NUM) s_grp[c][pos] = base + il;
                }
                cnt += __popc(mask);
            }
            if (lane == 0) s_cnt[c] = cnt;
        }
        __syncthreads();

        if (tid == 0) {
            int cmin = s_cnt[0];
            #pragma unroll
            for (int c = 1; c < 16; ++c) cmin = min(cmin, s_cnt[c]);
            s_done = (cmin >= GROUP_NUM);
        }
        __syncthreads();
        if (s_done) break;                          // uniform across block
    }
    __syncthreads();

    // Replicate-fill short groups with their first neighbor (the centroid
    // itself is always within radius of itself, so cnt >= 1).
    for (int k = tid; k < 16 * GROUP_NUM; k += 256) {
        const int c   = k >> 5;
        const int g   = k & 31;
        const int cnt = min(s_cnt[c], GROUP_NUM);
        if (g >= cnt) s_grp[c][g] = (cnt > 0) ? s_grp[c][0] : 0;
    }
    __syncthreads();

    // Emit [16, G+1, 3]: slot 0 = centroid xyz, slots 1..G = gathered - centroid
    float* outb = out + ((size_t)b * FPS_NUM + (size_t)tile * 16) * (GROUP_NUM + 1) * 3;
    for (int k = tid; k < 16 * (GROUP_NUM + 1); k += 256) {
        const int c = k / (GROUP_NUM + 1);
        const int g = k % (GROUP_NUM + 1);
        const float cx = s_c4[0][c], cy = s_c4[1][c], cz = s_c4[2][c];
        float* o = outb + ((size_t)c * (GROUP_NUM + 1) + g) * 3;
        if (g == 0) {
            o[0] = cx; o[1] = cy; o[2] = cz;
        } else {
            const int pi = s_grp[c][g - 1];
            o[0] = p[pi * 3 + 0] - cx;
            o[1] = p[pi * 3 + 1] - cy;
            o[2] = p[pi * 3 + 2] - cz;
        }
    }
}

extern "C" void kernel_launch(void* const* d_in, const int* in_sizes, int n_in,
                              void* d_out, int out_size, void* d_ws, size_t ws_size,
                              hipStream_t stream) {
    const float* xyz           = (const float*)d_in[0];
    const int*   farthest_init = (const int*)d_in[1];
    float*       out           = (float*)d_out;
    // d_out = combined_points [B,S,G+1,3] (405504 floats) ++ centroids_xyz [B,S,3]
    float* out_centroids = out + (size_t)NBATCH * FPS_NUM * (GROUP_NUM + 1) * 3;

    fps_kernel<<<NBATCH, 1024, 0, stream>>>(xyz, farthest_init, out_centroids);

    dim3 grid(FPS_NUM / 16, NBATCH);
    ballquery_group_kernel<<<grid, 256, 0, stream>>>(xyz, out_centroids, out);
}